// Embed_32779190403130
// MI455X (gfx1250) — compile-verified
//
#include <hip/hip_runtime.h>

// Native 4-wide float vector for __builtin_nontemporal_store (HIP's float4
// is a class type the builtin rejects).
typedef float v4f __attribute__((ext_vector_type(4)));

// Problem shape (fixed by the reference harness).
constexpr int B  = 4;
constexpr int S  = 2048;   // power of two -> modulo is a mask
constexpr int H  = 1024;
constexpr int DH = 4;
constexpr int WPB = 8;                 // waves per block (wave32 -> 256 threads)
constexpr float SQRT_H = 32.0f;        // sqrt(1024)
constexpr float LN_EPS = 1e-6f;
// ln(10000)/512 : pos freq = exp(-(h&511) * KFREQ)
constexpr float KFREQ = 0.017988946f;

__global__ __launch_bounds__(WPB * 32)
void embed_roll_ln_kernel(const int* __restrict__ y,
                          const int* __restrict__ hist,
                          const int* __restrict__ ym,
                          const float* __restrict__ toks,
                          const float* __restrict__ meta,
                          const float* __restrict__ gamma,
                          const float* __restrict__ beta,
                          float* __restrict__ out)
{
    __shared__ float lds[WPB * H];     // 32 KB: one 4KB token row per wave

    const int wave = threadIdx.x >> 5;
    const int lane = threadIdx.x & 31;
    const int row  = blockIdx.x * WPB + wave;   // [0, B*S)
    const int b    = row >> 11;                 // row / S  (S = 2048)
    const int s    = row & (S - 1);

    const int tok = y[row];
    const int mm  = ym[row];

    // ---- Stage 1: fire async DMA of the gathered token row into LDS -------
    // (all 32 lanes active -> 8 x 512B = 4KB per wave, tracked on ASYNCcnt;
    //  async loads complete IN ORDER, which stage 3 exploits)
    {
        const float* trow = toks + (size_t)tok * H;
        const unsigned ldsbase = (unsigned)(size_t)(&lds[wave * H]); // LDS byte offset
        #pragma unroll
        for (int c = 0; c < 8; ++c) {
            const float* g = trow + c * 128 + lane * 4;      // 16B per lane
            unsigned l = ldsbase + (unsigned)((c * 128 + lane * 4) * 4);
            asm volatile("global_load_async_to_lds_b128 %0, %1, off"
                         :: "v"(l), "v"(g) : "memory");
        }
    }

    // ---- Stage 2 (overlaps DMA): positional terms + meta gather ----------
    // net positional effect = circular left roll by sum(hist[b,:]) mod S
    const int shift =
        (hist[b * DH + 0] + hist[b * DH + 1] +
         hist[b * DH + 2] + hist[b * DH + 3]) & (S - 1);
    const float srot = (float)((s + shift) & (S - 1));

    const float* mrow = meta + (size_t)mm * H;

    float ev[32];
    #pragma unroll
    for (int c = 0; c < 8; ++c) {
        const int base = c * 128 + lane * 4;
        const float4 m4 = *(const float4*)(mrow + base);
        const float ma[4] = {m4.x, m4.y, m4.z, m4.w};
        #pragma unroll
        for (int k = 0; k < 4; ++k) {
            const int h = base + k;
            const float freq = __expf(-(float)(h & 511) * KFREQ);
            const float ang  = srot * freq;
            // chunks 0..3 -> h < 512 -> sin half; chunks 4..7 -> cos half
            const float p = (c < 4) ? __sinf(ang) : __cosf(ang);
            ev[c * 4 + k] = ma[k] + p;
        }
    }

    // ---- Stage 3: staged DMA drain + fuse + LayerNorm moments ------------
    // Chunk c is complete once ASYNCcnt <= 7-c (in-order completion), so the
    // fuse of chunk 0 overlaps the tail of the DMA for chunks 1..7.
    const float* lrow = &lds[wave * H];
    float sum = 0.0f, sumsq = 0.0f;
    #pragma unroll
    for (int c = 0; c < 8; ++c) {
        asm volatile("s_wait_asynccnt %0" :: "n"(7 - c) : "memory");
        const int base = c * 128 + lane * 4;
        const float4 t4 = *(const float4*)(lrow + base);     // ds_load_b128
        const float ta[4] = {t4.x, t4.y, t4.z, t4.w};
        #pragma unroll
        for (int k = 0; k < 4; ++k) {
            const float e = (ta[k] + ev[c * 4 + k]) * SQRT_H;
            ev[c * 4 + k] = e;
            sum   += e;
            sumsq += e * e;
        }
    }

    // ---- wave32 butterfly reduction (5 steps) ----------------------------
    #pragma unroll
    for (int m = 16; m >= 1; m >>= 1) {
        sum   += __shfl_xor(sum,   m, 32);
        sumsq += __shfl_xor(sumsq, m, 32);
    }
    const float mean = sum * (1.0f / H);
    const float var  = sumsq * (1.0f / H) - mean * mean;
    const float rstd = rsqrtf(var + LN_EPS);

    // ---- normalize + affine; NT 128-bit stores so the 32 MiB output ------
    // stream does not evict the (L2-resident) 128 MiB gather table.
    float* orow = out + (size_t)row * H;
    #pragma unroll
    for (int c = 0; c < 8; ++c) {
        const int base = c * 128 + lane * 4;
        const float4 g4 = *(const float4*)(gamma + base);
        const float4 b4 = *(const float4*)(beta  + base);
        v4f o;
        o.x = (ev[c * 4 + 0] - mean) * rstd * g4.x + b4.x;
        o.y = (ev[c * 4 + 1] - mean) * rstd * g4.y + b4.y;
        o.z = (ev[c * 4 + 2] - mean) * rstd * g4.z + b4.z;
        o.w = (ev[c * 4 + 3] - mean) * rstd * g4.w + b4.w;
        __builtin_nontemporal_store(o, (v4f*)(orow + base));
    }

    // ---- second tuple output: hist[:,0] as int32 bits past the floats ----
    if (blockIdx.x == 0 && threadIdx.x < B) {
        int* tail = (int*)(out + (size_t)B * S * H);
        tail[threadIdx.x] = hist[threadIdx.x * DH];
    }
}

extern "C" void kernel_launch(void* const* d_in, const int* in_sizes, int n_in,
                              void* d_out, int out_size, void* d_ws, size_t ws_size,
                              hipStream_t stream)
{
    const int*   y     = (const int*)  d_in[0];
    const int*   hist  = (const int*)  d_in[1];
    const int*   ym    = (const int*)  d_in[2];
    const float* toks  = (const float*)d_in[3];
    const float* meta  = (const float*)d_in[4];
    const float* gamma = (const float*)d_in[5];
    const float* beta  = (const float*)d_in[6];
    float* out = (float*)d_out;

    const dim3 grid((B * S) / WPB);   // 1024 blocks
    const dim3 block(WPB * 32);       // 8 wave32 per block
    hipLaunchKernelGGL(embed_roll_ln_kernel, grid, block, 0, stream,
                       y, hist, ym, toks, meta, gamma, beta, out);
}